// AttentionAggregator_50491635532351
// MI455X (gfx1250) — compile-verified
//
#include <hip/hip_runtime.h>
#include <math.h>

#define DD 128   // feature dim
#define KK 32    // neighbors per node
#define HH 32    // attention hidden
#define OO 128   // output dim per half

typedef __attribute__((ext_vector_type(2))) float v2f;
typedef __attribute__((ext_vector_type(8))) float v8f;

// D = A(16x4 f32) * B(4x16 f32) + C(16x16 f32)
__device__ __forceinline__ v8f wmma4(v2f a, v2f b, v8f c) {
  return __builtin_amdgcn_wmma_f32_16x16x4_f32(false, a, false, b, (short)0, c,
                                               false, false);
}

// ---------------------------------------------------------------------------
// Kernel 1: per-node attention + softmax + weighted neighbor aggregation.
// One wave32 per node; 4 waves per block; persistent wave-stride over nodes.
// Writes agg[N,128] to workspace.
// ---------------------------------------------------------------------------
__global__ __launch_bounds__(128)
void attn_agg_kernel(const float* __restrict__ node_feats,
                     const float* __restrict__ neib_feats,
                     const float* __restrict__ W1,   // [128,32]
                     const float* __restrict__ W2,   // [32,32]
                     float* __restrict__ agg_out,    // [N,128]
                     int N)
{
  __shared__ __align__(16) float sW1[DD * HH];       // 16 KB
  __shared__ __align__(16) float sW2[HH * HH];       //  4 KB
  __shared__ __align__(16) float sNeib[4][KK * DD];  // 64 KB (wave-private 16K)
  __shared__ __align__(16) float sX[4][DD];          //  2 KB
  __shared__ float sT[4][KK * 33];                   // padded, ~17 KB
  __shared__ float sVa[4][HH];
  __shared__ float sVb[4][HH];

  const int tid  = threadIdx.x;
  const int wave = tid >> 5;
  const int lane = tid & 31;
  const int lmod = lane & 15;
  const int lhi  = lane >> 4;   // 0 or 1

  for (int i = tid; i < DD * HH; i += 128) sW1[i] = W1[i];
  for (int i = tid; i < HH * HH; i += 128) sW2[i] = W2[i];
  __syncthreads();

  const int waves_total = gridDim.x * 4;
  for (int n = blockIdx.x * 4 + wave; n < N; n += waves_total) {
    // ---- prefetch NEXT node's 16KB neighbor tile + node row into cache ----
    // (overlaps this node's WMMA/tanh/softmax with the next HBM fetch)
    {
      const int nn = n + waves_total;
      if (nn < N) {
        const char* pn = (const char*)(neib_feats + (size_t)nn * KK * DD);
        #pragma unroll
        for (int i = 0; i < 4; ++i)           // 32 lanes * 128B * 4 = 16 KB
          __builtin_prefetch(pn + i * 4096 + lane * 128, 0, 3);
        const char* px = (const char*)(node_feats + (size_t)nn * DD);
        __builtin_prefetch(px + (lane & 3) * 128, 0, 3);  // 512B node row
      }
    }

    // ---- stage node row (coalesced float4) ----
    float4 xr = ((const float4*)(node_feats + (size_t)n * DD))[lane];
    ((float4*)sX[wave])[lane] = xr;

    // ---- node_t = tanh(x @ W1); lane h ----
    float acc = 0.f;
    #pragma unroll 4
    for (int d = 0; d < DD; ++d)
      acc = fmaf(sX[wave][d], sW1[d * HH + lane], acc);
    sVa[wave][lane] = tanhf(acc);

    // ---- node_att = node_t @ W2; lane h ----
    acc = 0.f;
    #pragma unroll
    for (int j = 0; j < HH; ++j)
      acc = fmaf(sVa[wave][j], sW2[j * HH + lane], acc);
    sVb[wave][lane] = acc;

    // ---- v = W2 @ node_att; lane j   (so scores = tanh(neib@W1) @ v) ----
    acc = 0.f;
    #pragma unroll
    for (int h = 0; h < HH; ++h)
      acc = fmaf(sW2[lane * HH + h], sVb[wave][h], acc);
    sVa[wave][lane] = acc;  // v

    // ---- stage 32x128 neighbor tile into wave-private LDS ----
    const float4* nb  = (const float4*)(neib_feats + (size_t)n * KK * DD);
    float4*       dst = (float4*)sNeib[wave];
    #pragma unroll 4
    for (int r = 0; r < KK; ++r)
      dst[r * (DD / 4) + lane] = nb[r * (DD / 4) + lane];

    // ---- T = tanh(neib @ W1) via f32 WMMA: 2 M-tiles x 2 N-tiles, K=128 ----
    v8f a00 = {}, a01 = {}, a10 = {}, a11 = {};
    const float* An = sNeib[wave];
    #pragma unroll 4
    for (int ks = 0; ks < DD / 4; ++ks) {
      const int kc = ks * 4 + 2 * lhi;
      // A frags: lane holds A[M=lmod(+16)][K=kc..kc+1] -> b64 LDS read
      v2f af0 = *(const v2f*)&An[(lmod     ) * DD + kc];
      v2f af1 = *(const v2f*)&An[(lmod + 16) * DD + kc];
      // B frags: lane holds B[K=kc+j][N=lmod(+16)]
      v2f bf0, bf1;
      bf0.x = sW1[(kc    ) * HH + lmod];
      bf0.y = sW1[(kc + 1) * HH + lmod];
      bf1.x = sW1[(kc    ) * HH + 16 + lmod];
      bf1.y = sW1[(kc + 1) * HH + 16 + lmod];
      a00 = wmma4(af0, bf0, a00);
      a01 = wmma4(af0, bf1, a01);
      a10 = wmma4(af1, bf0, a10);
      a11 = wmma4(af1, bf1, a11);
    }
    // tanh + spill T to LDS (row-padded stride 33 => conflict-free col reads)
    #pragma unroll
    for (int r = 0; r < 8; ++r) {
      const int row = r + 8 * lhi;
      sT[wave][(row     ) * 33 + lmod     ] = tanhf(a00[r]);
      sT[wave][(row     ) * 33 + lmod + 16] = tanhf(a01[r]);
      sT[wave][(row + 16) * 33 + lmod     ] = tanhf(a10[r]);
      sT[wave][(row + 16) * 33 + lmod + 16] = tanhf(a11[r]);
    }

    // ---- scores[k] = T[k,:] . v  (lane k) ----
    float sc = 0.f;
    #pragma unroll
    for (int h = 0; h < HH; ++h)
      sc = fmaf(sT[wave][lane * 33 + h], sVa[wave][h], sc);

    // ---- softmax over 32 lanes (wave32) ----
    float m = sc;
    #pragma unroll
    for (int off = 16; off; off >>= 1)
      m = fmaxf(m, __shfl_xor(m, off, 32));
    const float e = expf(sc - m);
    float s = e;
    #pragma unroll
    for (int off = 16; off; off >>= 1)
      s += __shfl_xor(s, off, 32);
    const float w = e / s;

    // ---- agg[d] = sum_k ws[k]*neib[k,d]; lane owns 4 columns ----
    float4 ag = make_float4(0.f, 0.f, 0.f, 0.f);
    #pragma unroll 4
    for (int k = 0; k < KK; ++k) {
      const float wk = __shfl(w, k, 32);  // v_readlane broadcast
      float4 nv = ((const float4*)sNeib[wave])[k * (DD / 4) + lane];
      ag.x = fmaf(wk, nv.x, ag.x);
      ag.y = fmaf(wk, nv.y, ag.y);
      ag.z = fmaf(wk, nv.z, ag.z);
      ag.w = fmaf(wk, nv.w, ag.w);
    }
    ((float4*)(agg_out + (size_t)n * DD))[lane] = ag;
  }
}

// ---------------------------------------------------------------------------
// Kernel 2: out = relu([node_feats @ W_node, agg @ W_neib]); [N,256].
// One block per 16-node tile; wave w owns 64 output columns.
// ---------------------------------------------------------------------------
__global__ __launch_bounds__(128)
void out_gemm_kernel(const float* __restrict__ node_feats,
                     const float* __restrict__ agg,
                     const float* __restrict__ Wn,  // [128,128]
                     const float* __restrict__ Wb,  // [128,128]
                     float* __restrict__ out,       // [N,256]
                     int N)
{
  __shared__ __align__(16) float sA[2][16 * DD];  // 16 KB
  const int tid  = threadIdx.x;
  const int wave = tid >> 5;
  const int lane = tid & 31;
  const int lmod = lane & 15;
  const int lhi  = lane >> 4;
  const int n0   = blockIdx.x * 16;
  if (n0 + 16 > N) return;  // N divisible by 16 for this problem

  {
    const float4* s0 = (const float4*)(node_feats + (size_t)n0 * DD);
    const float4* s1 = (const float4*)(agg        + (size_t)n0 * DD);
    float4* d0 = (float4*)sA[0];
    float4* d1 = (float4*)sA[1];
    for (int i = tid; i < 16 * (DD / 4); i += 128) { d0[i] = s0[i]; d1[i] = s1[i]; }
  }
  __syncthreads();

  const float* A = sA[wave >> 1];            // waves 0,1 -> node; 2,3 -> agg
  const float* W = (wave < 2) ? Wn : Wb;
  const int colBase = (wave & 1) * 64;

  v8f acc0 = {}, acc1 = {}, acc2 = {}, acc3 = {};
  #pragma unroll 4
  for (int ks = 0; ks < DD / 4; ++ks) {
    const int kc = ks * 4 + 2 * lhi;
    v2f a = *(const v2f*)&A[lmod * DD + kc];
    v2f b0, b1, b2, b3;
    b0.x = W[(size_t)(kc    ) * OO + colBase      + lmod];
    b0.y = W[(size_t)(kc + 1) * OO + colBase      + lmod];
    b1.x = W[(size_t)(kc    ) * OO + colBase + 16 + lmod];
    b1.y = W[(size_t)(kc + 1) * OO + colBase + 16 + lmod];
    b2.x = W[(size_t)(kc    ) * OO + colBase + 32 + lmod];
    b2.y = W[(size_t)(kc + 1) * OO + colBase + 32 + lmod];
    b3.x = W[(size_t)(kc    ) * OO + colBase + 48 + lmod];
    b3.y = W[(size_t)(kc + 1) * OO + colBase + 48 + lmod];
    acc0 = wmma4(a, b0, acc0);
    acc1 = wmma4(a, b1, acc1);
    acc2 = wmma4(a, b2, acc2);
    acc3 = wmma4(a, b3, acc3);
  }

  const int outColBase = ((wave < 2) ? 0 : 128) + (wave & 1) * 64;
  #pragma unroll
  for (int r = 0; r < 8; ++r) {
    const int row = n0 + r + 8 * lhi;
    float* orow = out + (size_t)row * 256 + outColBase + lmod;
    orow[0]  = fmaxf(acc0[r], 0.f);
    orow[16] = fmaxf(acc1[r], 0.f);
    orow[32] = fmaxf(acc2[r], 0.f);
    orow[48] = fmaxf(acc3[r], 0.f);
  }
}

extern "C" void kernel_launch(void* const* d_in, const int* in_sizes, int n_in,
                              void* d_out, int out_size, void* d_ws, size_t ws_size,
                              hipStream_t stream)
{
  const float* node_feats = (const float*)d_in[0];
  const float* neib_feats = (const float*)d_in[1];
  // d_in[2] (node_ids), d_in[3] (neib_ids): unused by the reference
  const float* W_att1 = (const float*)d_in[4];
  const float* W_att2 = (const float*)d_in[5];
  const float* W_node = (const float*)d_in[6];
  const float* W_neib = (const float*)d_in[7];
  float* out = (float*)d_out;
  float* agg = (float*)d_ws;  // N*128 floats = 25.6 MB scratch

  const int N = in_sizes[0] / DD;

  attn_agg_kernel<<<1024, 128, 0, stream>>>(node_feats, neib_feats,
                                            W_att1, W_att2, agg, N);
  out_gemm_kernel<<<(N + 15) / 16, 128, 0, stream>>>(node_feats, agg,
                                                     W_node, W_neib, out, N);
}